// PhoenixRetrievalModel_83056077570806
// MI455X (gfx1250) — compile-verified
//
#include <hip/hip_runtime.h>
#include <math.h>

typedef __attribute__((ext_vector_type(2))) float f32x2;
typedef __attribute__((ext_vector_type(8))) float f32x8;
typedef __attribute__((ext_vector_type(4))) unsigned u32x4;
typedef __attribute__((ext_vector_type(8))) int i32x8;
typedef __attribute__((ext_vector_type(4))) int i32x4;

#define B_    64
#define S_    160
#define D_    512
#define H_    8
#define DH_   64
#define DFF_  2048
#define NC_   500000
#define TOPK_ 256
#define BS_   (B_ * S_)
#define BINS  2048
#define CAP   4096
#define LDS_ROW 132  // 128 data floats + 4 pad floats (bank-conflict-free)

static __device__ __forceinline__ f32x8 wmma_f32(f32x2 a, f32x2 b, f32x8 c) {
  // D = A(16x4) * B(4x16) + C, fp32 end-to-end
  return __builtin_amdgcn_wmma_f32_16x16x4_f32(false, a, false, b, (short)0, c, false, false);
}

static __device__ __forceinline__ float waveReduceSum(float v) {
  for (int off = 16; off > 0; off >>= 1) v += __shfl_xor(v, off, 32);
  return v;
}
static __device__ __forceinline__ float waveReduceMax(float v) {
  for (int off = 16; off > 0; off >>= 1) v = fmaxf(v, __shfl_xor(v, off, 32));
  return v;
}

// monotone float -> uint key (larger float => larger uint)
static __device__ __forceinline__ unsigned fkey(float f) {
  unsigned u = __float_as_uint(f);
  return (u & 0x80000000u) ? ~u : (u | 0x80000000u);
}

// ------------------------------------------------------------- TDM 2D load
// Issue TENSOR_LOAD_TO_LDS of a (tile_d1 rows x tile_d0 elems) fp32 tile.
// Hardware pad: +4 DWORDs after every 128 DWORDs -> LDS row stride 132 floats.
static __device__ __forceinline__ void tdm_load_2d(unsigned lds_off, const float* gaddr,
                                                   unsigned tensor_d0, unsigned tensor_d1,
                                                   unsigned tile_d0, unsigned tile_d1,
                                                   unsigned stride0_elems) {
  unsigned long long ga = (unsigned long long)(size_t)gaddr;
  u32x4 g0;
  g0[0] = 1u;                                    // count=1, user mode, no gather
  g0[1] = lds_off;                               // lds_addr (bytes)
  g0[2] = (unsigned)(ga & 0xFFFFFFFFu);          // global_addr[31:0]
  g0[3] = (unsigned)((ga >> 32) & 0x01FFFFFFu)   // global_addr[56:32]
          | (2u << 30);                          // type = 2 ("image")
  i32x8 g1;
  g1[0] = (int)((2u << 16)     // data_size = 4 bytes
              | (1u << 20)     // pad_enable
              | (6u << 22)     // pad_interval: every 128 DWORDs
              | (3u << 25));   // pad_amount:  4 DWORDs
  g1[1] = (int)((tensor_d0 & 0xFFFFu) << 16);                    // tensor_dim0[15:0]
  g1[2] = (int)((tensor_d0 >> 16) | ((tensor_d1 & 0xFFFFu) << 16));
  g1[3] = (int)((tensor_d1 >> 16) | (tile_d0 << 16));            // tile_dim0
  g1[4] = (int)(tile_d1 & 0xFFFFu);                              // tile_dim1 (tile_dim2=0)
  g1[5] = (int)stride0_elems;                                    // tensor_dim0_stride[31:0]
  g1[6] = 0;                                                     // stride hi / dim1_stride
  g1[7] = 0;
  i32x4 z4 = {0, 0, 0, 0};
  i32x8 z8 = {0, 0, 0, 0, 0, 0, 0, 0};
  // 6-arg toolchain variant: (g0, g1, g2, g3, extra, cpol)
  __builtin_amdgcn_tensor_load_to_lds(g0, g1, z4, z4, z8, 0);
}

// ---------------------------------------------------------------- RMS norm
__global__ __launch_bounds__(256) void rms_kernel(const float* __restrict__ in,
                                                  float* __restrict__ out, int rows) {
  int wave = (blockIdx.x * blockDim.x + threadIdx.x) >> 5;
  int lane = threadIdx.x & 31;
  if (wave >= rows) return;
  const float* p = in + (size_t)wave * D_;
  float r[16];
  float ss = 0.f;
#pragma unroll
  for (int i = 0; i < 16; ++i) {
    r[i] = p[lane + i * 32];
    ss += r[i] * r[i];
  }
  ss = waveReduceSum(ss);
  float sc = rsqrtf(ss * (1.0f / D_) + 1e-6f);
  float* q = out + (size_t)wave * D_;
#pragma unroll
  for (int i = 0; i < 16; ++i) q[lane + i * 32] = r[i] * sc;
}

// ------------------------------------------------- generic fp32 WMMA GEMM
// Each wave computes a 16x64 tile (4 accumulators): A-frag reused 4x.
// MODE 0: store, 1: +R residual, 2: SiLU
template <int MODE>
__global__ __launch_bounds__(256) void gemm_kernel(const float* __restrict__ A,
                                                   const float* __restrict__ Bm,
                                                   const float* __restrict__ R,
                                                   float* __restrict__ C,
                                                   int M, int N, int K) {
  int wave = (blockIdx.x * blockDim.x + threadIdx.x) >> 5;
  int lane = threadIdx.x & 31;
  int tn_cnt = N >> 6;
  int total = (M >> 4) * tn_cnt;
  if (wave >= total) return;  // wave-uniform: EXEC stays all-ones for WMMA
  int tm = wave / tn_cnt, tn = wave - tm * tn_cnt;
  int half = lane >> 4, l = lane & 15;
  int row = tm * 16 + l;
  int col0 = tn * 64 + l;
  const float* ap = A + (size_t)row * K + 2 * half;
  const float* bbase = Bm + (size_t)(2 * half) * N + col0;
  f32x8 acc0 = {0, 0, 0, 0, 0, 0, 0, 0};
  f32x8 acc1 = acc0, acc2 = acc0, acc3 = acc0;
  for (int k = 0; k < K; k += 4) {
    float2 av = *(const float2*)(ap + k);
    f32x2 a;
    a.x = av.x;
    a.y = av.y;
    const float* bp = bbase + (size_t)k * N;
    f32x2 b0, b1, b2, b3;
    b0.x = bp[0];
    b0.y = bp[N];
    b1.x = bp[16];
    b1.y = bp[N + 16];
    b2.x = bp[32];
    b2.y = bp[N + 32];
    b3.x = bp[48];
    b3.y = bp[N + 48];
    acc0 = wmma_f32(a, b0, acc0);
    acc1 = wmma_f32(a, b1, acc1);
    acc2 = wmma_f32(a, b2, acc2);
    acc3 = wmma_f32(a, b3, acc3);
  }
#pragma unroll
  for (int j = 0; j < 8; ++j) {
    int r = tm * 16 + j + 8 * half;
    float vv[4] = {acc0[j], acc1[j], acc2[j], acc3[j]};
#pragma unroll
    for (int g = 0; g < 4; ++g) {
      int col = col0 + g * 16;
      float v = vv[g];
      if (MODE == 1) v += R[(size_t)r * N + col];
      if (MODE == 2) v = v / (1.0f + __expf(-v));  // SiLU
      C[(size_t)r * N + col] = v;
    }
  }
}

// ---------------------------------------------------------------- attention
// one block per (b, h, 32-query tile); scores staged in LDS
__global__ __launch_bounds__(256) void attn_kernel(const float* __restrict__ q,
                                                   const float* __restrict__ k,
                                                   const float* __restrict__ v,
                                                   const int* __restrict__ pmask,
                                                   float* __restrict__ o) {
  __shared__ float sc[32 * S_];
  const int tid = threadIdx.x;
  int bid = blockIdx.x;
  int qt = bid % (S_ / 32);
  bid /= (S_ / 32);
  int h = bid % H_;
  int b = bid / H_;
  const size_t base = ((size_t)b * S_) * D_ + (size_t)h * DH_;
  const int q0 = qt * 32;

  for (int idx = tid; idx < 32 * S_; idx += 256) {
    int qi = idx / S_, kj = idx % S_;
    const float* qp = q + base + (size_t)(q0 + qi) * D_;
    const float* kp = k + base + (size_t)kj * D_;
    float s = 0.f;
#pragma unroll 8
    for (int d = 0; d < DH_; ++d) s += qp[d] * kp[d];
    s *= 0.125f;  // Dh^-0.5
    s += (pmask[b * S_ + kj] > 0) ? 0.f : -1e12f;
    sc[qi * S_ + kj] = s;
  }
  __syncthreads();

  int wave = tid >> 5, lane = tid & 31;
  for (int r = wave; r < 32; r += 8) {
    float m = -3.4e38f;
    for (int j = lane; j < S_; j += 32) m = fmaxf(m, sc[r * S_ + j]);
    m = waveReduceMax(m);
    float ssum = 0.f;
    for (int j = lane; j < S_; j += 32) {
      float e = __expf(sc[r * S_ + j] - m);
      sc[r * S_ + j] = e;
      ssum += e;
    }
    ssum = waveReduceSum(ssum);
    float inv = 1.0f / ssum;
    for (int j = lane; j < S_; j += 32) sc[r * S_ + j] *= inv;
  }
  __syncthreads();

  for (int idx = tid; idx < 32 * DH_; idx += 256) {
    int qi = idx / DH_, d = idx % DH_;
    float s = 0.f;
    for (int kj = 0; kj < S_; ++kj) s += sc[qi * S_ + kj] * v[base + (size_t)kj * D_ + d];
    o[base + (size_t)(q0 + qi) * D_ + d] = s;
  }
}

// ------------------------------------------------ masked mean pool + L2 norm
__global__ __launch_bounds__(256) void pool_kernel(const float* __restrict__ x,
                                                   const int* __restrict__ pmask,
                                                   float* __restrict__ user) {
  __shared__ float red[256];
  __shared__ float upre[D_];
  __shared__ float s_norm;
  int b = blockIdx.x, tid = threadIdx.x;
  float msum = 0.f;
  for (int s = 0; s < S_; ++s) msum += (pmask[b * S_ + s] > 0) ? 1.f : 0.f;
  float denom = fmaxf(msum, 1.f);
  float ss = 0.f;
  for (int d = tid; d < D_; d += 256) {
    float acc = 0.f;
    for (int s = 0; s < S_; ++s) {
      float m = (pmask[b * S_ + s] > 0) ? 1.f : 0.f;
      acc += x[((size_t)b * S_ + s) * D_ + d] * m;
    }
    float u = acc / denom;
    upre[d] = u;
    ss += u * u;
  }
  red[tid] = ss;
  __syncthreads();
  for (int off = 128; off > 0; off >>= 1) {
    if (tid < off) red[tid] += red[tid + off];
    __syncthreads();
  }
  if (tid == 0) s_norm = sqrtf(fmaxf(red[0], 1e-12f));
  __syncthreads();
  float inv = 1.0f / s_norm;
  for (int d = tid; d < D_; d += 256) user[(size_t)b * D_ + d] = upre[d] * inv;
}

// ----------------------------------------- similarity: user @ corpus^T (WMMA)
// block: 32 corpus rows x 64 users; corpus tiles streamed by the Tensor Data
// Mover into double-buffered LDS (TDM pad recreates the 132-float stride).
__global__ __launch_bounds__(256) void sim_kernel(const float* __restrict__ user,
                                                  const float* __restrict__ corpus,
                                                  const int* __restrict__ cmask,
                                                  float* __restrict__ sims) {
  __shared__ float Bsh[2][32 * LDS_ROW];
  const unsigned kBufBytes = 32 * LDS_ROW * 4;
  // LDS aperture maps addr[31:0] directly to the LDS byte offset (ISA 10.2)
  const unsigned ldsBase = (unsigned)(size_t)(&Bsh[0][0]);
  const int tid = threadIdx.x;
  const int lane = tid & 31;
  const int wave = tid >> 5;
  const int half = lane >> 4;
  const int l = lane & 15;
  const int n0 = blockIdx.x * 32;
  const int mt = wave & 3, nt = wave >> 2;
  const int m0 = mt * 16;
  const int colL = nt * 16 + l;
  f32x8 acc = {0, 0, 0, 0, 0, 0, 0, 0};

  // prime the pipeline: DMA chunk 0 into buffer 0 (wave 0 only; TDM ignores EXEC)
  if (wave == 0) {
    tdm_load_2d(ldsBase, corpus + (size_t)n0 * D_,
                /*tensor_d0=*/D_, /*tensor_d1=*/32,
                /*tile_d0=*/128, /*tile_d1=*/32, /*stride0=*/D_);
  }

  for (int kci = 0; kci < 4; ++kci) {
    const int kc = kci * 128;
    if (wave == 0) __builtin_amdgcn_s_wait_tensorcnt(0);
    __syncthreads();  // chunk kci resident for everyone
    // kick off next chunk into the other buffer while we compute
    if (wave == 0 && kci + 1 < 4) {
      tdm_load_2d(ldsBase + ((kci + 1) & 1) * kBufBytes,
                  corpus + (size_t)n0 * D_ + kc + 128, D_, 32, 128, 32, D_);
    }
    const float* bufp = &Bsh[kci & 1][0];
    // DMA wrote Bsh behind the optimizer's back: escape the pointer + memory
    // clobber so the ds_loads below are kept and ordered after the wait.
    asm volatile("" : : "v"(bufp) : "memory");
    const float* Arow = user + (size_t)(m0 + l) * D_ + kc + 2 * half;
#pragma unroll
    for (int ks = 0; ks < 128; ks += 4) {
      float2 av = *(const float2*)(Arow + ks);
      f32x2 a;
      a.x = av.x;
      a.y = av.y;
      const float* bp = bufp + colL * LDS_ROW + ks + 2 * half;
      f32x2 b;
      b.x = bp[0];
      b.y = bp[1];
      acc = wmma_f32(a, b, acc);
    }
    __syncthreads();  // all reads of this buffer done before it is re-filled
  }

  const int col = n0 + colL;
  const bool ok = cmask[col] > 0;
#pragma unroll
  for (int j = 0; j < 8; ++j) {
    int r = m0 + j + 8 * half;
    float sv = ok ? acc[j] : -1e12f;
    sims[(size_t)r * NC_ + col] = sv;
  }
}

// --------------------------------- top-k: histogram select + bitonic (LDS)
__global__ __launch_bounds__(256) void topk_kernel(const float* __restrict__ sims,
                                                   float* __restrict__ outScore,
                                                   int* __restrict__ outIdx) {
  __shared__ unsigned hist[BINS];
  __shared__ float sKey[CAP];
  __shared__ int sIdx[CAP];
  __shared__ int s_pivot, s_cnt, s_c, s_M;
  int row = blockIdx.x, tid = threadIdx.x;
  for (int i = tid; i < BINS; i += 256) hist[i] = 0;
  if (tid == 0) s_cnt = 0;
  __syncthreads();

  const float* Sr = sims + (size_t)row * NC_;
  for (int i = tid; i < NC_; i += 256) {
    unsigned u = fkey(Sr[i]);
    atomicAdd(&hist[u >> 21], 1u);
  }
  __syncthreads();

  if (tid == 0) {
    unsigned cum = 0;
    int p = 0;
    for (int bi = BINS - 1; bi >= 0; --bi) {
      cum += hist[bi];
      if (cum >= TOPK_) { p = bi; break; }
    }
    s_pivot = p;
  }
  __syncthreads();

  int pivot = s_pivot;
  for (int i = tid; i < NC_; i += 256) {
    float xv = Sr[i];
    unsigned u = fkey(xv);
    if ((int)(u >> 21) >= pivot) {
      int pos = atomicAdd(&s_cnt, 1);
      if (pos < CAP) {
        sKey[pos] = xv;
        sIdx[pos] = i;
      }
    }
  }
  __syncthreads();

  if (tid == 0) {
    int c = s_cnt;
    if (c > CAP) c = CAP;
    int m = 256;
    while (m < c) m <<= 1;
    s_c = c;
    s_M = m;
  }
  __syncthreads();
  int c = s_c, M = s_M;
  for (int i = c + tid; i < M; i += 256) {
    sKey[i] = -3.4e38f;
    sIdx[i] = -1;
  }
  __syncthreads();

  // bitonic sort, descending
  for (int ksz = 2; ksz <= M; ksz <<= 1) {
    for (int j = ksz >> 1; j > 0; j >>= 1) {
      for (int i = tid; i < M; i += 256) {
        int ixj = i ^ j;
        if (ixj > i) {
          float a = sKey[i], bv = sKey[ixj];
          bool desc = ((i & ksz) == 0);
          if (desc ? (a < bv) : (a > bv)) {
            sKey[i] = bv;
            sKey[ixj] = a;
            int t = sIdx[i];
            sIdx[i] = sIdx[ixj];
            sIdx[ixj] = t;
          }
        }
      }
      __syncthreads();
    }
  }
  if (tid < TOPK_) {
    outScore[row * TOPK_ + tid] = sKey[tid];
    outIdx[row * TOPK_ + tid] = sIdx[tid];
  }
}

// --------------------------------------------------------------------------
extern "C" void kernel_launch(void* const* d_in, const int* in_sizes, int n_in,
                              void* d_out, int out_size, void* d_ws, size_t ws_size,
                              hipStream_t stream) {
  (void)in_sizes; (void)n_in; (void)out_size; (void)ws_size;
  const float* emb    = (const float*)d_in[0];
  const int*   pmask  = (const int*)d_in[1];
  const float* corpus = (const float*)d_in[2];
  const int*   cmask  = (const int*)d_in[3];
  const float* Wq = (const float*)d_in[4];
  const float* Wk = (const float*)d_in[5];
  const float* Wv = (const float*)d_in[6];
  const float* Wo = (const float*)d_in[7];
  const float* W1 = (const float*)d_in[8];
  const float* W2 = (const float*)d_in[9];

  float* ws = (float*)d_ws;
  const size_t SZ = (size_t)BS_ * D_;
  float* b0 = ws;            // h, later o
  float* b1 = ws + SZ;       // q, later x
  float* b2 = ws + 2 * SZ;   // k, later y = rms(x)
  float* b3 = ws + 3 * SZ;   // v, later x2 (final)
  float* big = ws + 4 * SZ;  // t (BS x 2048), later sims (64 x NC)

  float* user_out = (float*)d_out;                         // (64,512)
  int*   idx_out  = (int*)d_out + B_ * D_;                 // (64,256) int32
  float* scr_out  = (float*)d_out + B_ * D_ + B_ * TOPK_;  // (64,256)

  dim3 blk(256);
  const int wavesQ = (BS_ / 16) * (D_ / 64);
  const int blocksQ = (wavesQ + 7) / 8;

  // 1. h = rms(emb)
  rms_kernel<<<dim3(BS_ / 8), blk, 0, stream>>>(emb, b0, BS_);
  // 2. q,k,v projections
  gemm_kernel<0><<<dim3(blocksQ), blk, 0, stream>>>(b0, Wq, nullptr, b1, BS_, D_, D_);
  gemm_kernel<0><<<dim3(blocksQ), blk, 0, stream>>>(b0, Wk, nullptr, b2, BS_, D_, D_);
  gemm_kernel<0><<<dim3(blocksQ), blk, 0, stream>>>(b0, Wv, nullptr, b3, BS_, D_, D_);
  // 3. attention -> o (reuse b0)
  attn_kernel<<<dim3(B_ * H_ * (S_ / 32)), blk, 0, stream>>>(b1, b2, b3, pmask, b0);
  // 4. x = emb + o @ Wo -> b1
  gemm_kernel<1><<<dim3(blocksQ), blk, 0, stream>>>(b0, Wo, emb, b1, BS_, D_, D_);
  // 5. y = rms(x) -> b2
  rms_kernel<<<dim3(BS_ / 8), blk, 0, stream>>>(b1, b2, BS_);
  // 6. t = silu(y @ W1) -> big
  const int wavesF = (BS_ / 16) * (DFF_ / 64);
  gemm_kernel<2><<<dim3((wavesF + 7) / 8), blk, 0, stream>>>(b2, W1, nullptr, big, BS_, DFF_, D_);
  // 7. x2 = x + t @ W2 -> b3
  gemm_kernel<1><<<dim3(blocksQ), blk, 0, stream>>>(big, W2, b1, b3, BS_, D_, DFF_);
  // 8. user = normalize(masked-mean(x2)) -> d_out
  pool_kernel<<<dim3(B_), blk, 0, stream>>>(b3, pmask, user_out);
  // 9. sims = user @ corpus^T (masked) -> big   [TDM-streamed corpus]
  sim_kernel<<<dim3(NC_ / 32), blk, 0, stream>>>(user_out, corpus, cmask, big);
  // 10. top-256 per row
  topk_kernel<<<dim3(B_), blk, 0, stream>>>(big, scr_out, idx_out);
}